// DeeperGCNfp_27410481283104
// MI455X (gfx1250) — compile-verified
//
#include <hip/hip_runtime.h>

typedef __attribute__((ext_vector_type(2))) float v2f;
typedef __attribute__((ext_vector_type(8))) float v8f;

#define DD 128

__device__ __forceinline__ void atomAddF(float* p, float v) {
  __hip_atomic_fetch_add(p, v, __ATOMIC_RELAXED, __HIP_MEMORY_SCOPE_AGENT);
}
__device__ __forceinline__ int atomAddI(int* p, int v) {
  return __hip_atomic_fetch_add(p, v, __ATOMIC_RELAXED, __HIP_MEMORY_SCOPE_AGENT);
}

// ---------------- embeddings ----------------
__global__ __launch_bounds__(256) void k_atom(const int* __restrict__ x,
                                              const float* __restrict__ aemb,
                                              float* __restrict__ h, long nd) {
  long idx = (long)blockIdx.x * blockDim.x + threadIdx.x;
  if (idx >= nd) return;
  long n = idx >> 7;
  int c = (int)(idx & (DD - 1));
  float s = 0.f;
#pragma unroll
  for (int f = 0; f < 9; ++f) {
    int id = x[n * 9 + f];
    s += aemb[((long)(f * 64 + id)) * DD + c];
  }
  h[idx] = s;
}

// ---------------- generic utility ----------------
__global__ __launch_bounds__(256) void k_zero(float* __restrict__ p, long n) {
  long i = (long)blockIdx.x * blockDim.x + threadIdx.x;
  if (i < n) p[i] = 0.f;
}
__global__ __launch_bounds__(256) void k_zero_int(int* __restrict__ p, long n) {
  long i = (long)blockIdx.x * blockDim.x + threadIdx.x;
  if (i < n) p[i] = 0;
}

// ---------------- CSR build (once per call; edge_index is constant across layers) -----
__global__ __launch_bounds__(256) void k_count(const int* __restrict__ ei, int* __restrict__ deg,
                                               int E) {
  long e = (long)blockIdx.x * blockDim.x + threadIdx.x;
  if (e >= E) return;
  atomAddI(&deg[ei[(long)E + e]], 1);
}

// single-block chunked Hillis-Steele scan -> exclusive rowptr[N+1]
__global__ __launch_bounds__(1024) void k_scan(const int* __restrict__ deg,
                                               int* __restrict__ rowptr, int N) {
  __shared__ int sh[1024];
  __shared__ int carry;
  int tid = threadIdx.x;
  if (tid == 0) {
    carry = 0;
    rowptr[0] = 0;
  }
  __syncthreads();
  for (int base = 0; base < N; base += 1024) {
    int i = base + tid;
    int v = (i < N) ? deg[i] : 0;
    sh[tid] = v;
    __syncthreads();
    for (int off = 1; off < 1024; off <<= 1) {
      int t = (tid >= off) ? sh[tid - off] : 0;
      __syncthreads();
      sh[tid] += t;
      __syncthreads();
    }
    if (i < N) rowptr[i + 1] = carry + sh[tid];
    __syncthreads();
    if (tid == 0) carry += sh[1023];
    __syncthreads();
  }
}

__global__ __launch_bounds__(256) void k_scatter(const int* __restrict__ ei,
                                                 const int* __restrict__ rowptr,
                                                 int* __restrict__ fill,
                                                 int* __restrict__ eidx, int E) {
  long e = (long)blockIdx.x * blockDim.x + threadIdx.x;
  if (e >= E) return;
  int d = ei[(long)E + e];
  int pos = atomAddI(&fill[d], 1);
  eidx[rowptr[d] + pos] = (int)e;
}

// ---------------- edge message helper (recompute bond emb, L0-resident tables) --------
__device__ __forceinline__ void edge_msg(const float* __restrict__ hin,
                                         const float* __restrict__ bemb,
                                         const int* __restrict__ ea, long e, int s, int c,
                                         float* mg) {
  const float4 hv = *(const float4*)(hin + (long)s * DD + c);
  int a0 = ea[e * 3 + 0], a1 = ea[e * 3 + 1], a2 = ea[e * 3 + 2];
  const float4 b0 = *(const float4*)(bemb + (long)a0 * DD + c);
  const float4 b1 = *(const float4*)(bemb + (long)(8 + a1) * DD + c);
  const float4 b2 = *(const float4*)(bemb + (long)(16 + a2) * DD + c);
  mg[0] = fmaxf(hv.x + b0.x + b1.x + b2.x, 0.f) + 1e-7f;
  mg[1] = fmaxf(hv.y + b0.y + b1.y + b2.y, 0.f) + 1e-7f;
  mg[2] = fmaxf(hv.z + b0.z + b1.z + b2.z, 0.f) + 1e-7f;
  mg[3] = fmaxf(hv.w + b0.w + b1.w + b2.w, 0.f) + 1e-7f;
}

// ---------------- gather aggregation: online softmax, wave per dst node --------------
// t[n] = hin[n] + sum_e alpha_e * msg_e   (alpha = softmax over incoming edges, per channel)
__global__ __launch_bounds__(256) void k_aggr(const int* __restrict__ ei,
                                              const int* __restrict__ ea,
                                              const int* __restrict__ rowptr,
                                              const int* __restrict__ eidx,
                                              const float* __restrict__ bemb,
                                              const float* __restrict__ hin,
                                              const float* __restrict__ ts, int l,
                                              float* __restrict__ t, int N, int E) {
  long n = ((long)blockIdx.x * blockDim.x + threadIdx.x) >> 5;
  int lane = threadIdx.x & 31;
  if (n >= N) return;
  int c = lane * 4;
  float tsc = ts[l];
  float m[4] = {-3.4e38f, -3.4e38f, -3.4e38f, -3.4e38f};
  float s[4] = {0.f, 0.f, 0.f, 0.f};
  float w[4] = {0.f, 0.f, 0.f, 0.f};
  int beg = rowptr[n], end = rowptr[n + 1];
  for (int i = beg; i < end; ++i) {
    int e = eidx[i];
    int src = ei[e];
    float mg[4];
    edge_msg(hin, bemb, ea, e, src, c, mg);
#pragma unroll
    for (int j = 0; j < 4; ++j) {
      float lg = mg[j] * tsc;
      float mn = fmaxf(m[j], lg);
      float scale = __expf(m[j] - mn);  // first iter: exp(-huge) == 0, s=w=0 anyway
      float p = __expf(lg - mn);
      s[j] = s[j] * scale + p;
      w[j] = w[j] * scale + p * mg[j];
      m[j] = mn;
    }
  }
  const float4 hv = *(const float4*)(hin + n * DD + c);
  float4 o;
  o.x = hv.x + w[0] / (s[0] + 1e-16f);
  o.y = hv.y + w[1] / (s[1] + 1e-16f);
  o.z = hv.z + w[2] / (s[2] + 1e-16f);
  o.w = hv.w + w[3] / (s[3] + 1e-16f);
  *(float4*)(t + n * DD + c) = o;
}

// ---------------- WMMA fp32 GEMM: t <- t @ w + b (+ resid), in-place ----------------
// block = 256 threads = 8 waves; each wave owns a 16-row tile; 128 rows/block.
__global__ __launch_bounds__(256) void k_gemm(float* __restrict__ t, const float* __restrict__ w,
                                              const float* __restrict__ b,
                                              const float* __restrict__ resid, int N) {
  __shared__ float sW[DD * DD];       // 64 KB weights
  __shared__ float sA[8 * 16 * 132];  // padded A tiles (stride 132 -> conflict-free)
  const int tid = threadIdx.x;
  const int wv = tid >> 5;
  const int lane = tid & 31;

  for (int i = tid; i < DD * DD / 4; i += 256) ((float4*)sW)[i] = ((const float4*)w)[i];

  const int rowbase = blockIdx.x * 128 + wv * 16;
  for (int i = lane; i < 16 * 32; i += 32) {
    int r = i >> 5;
    int k4 = i & 31;
    int row = rowbase + r;
    if (row >= N) row = N - 1;  // clamp stays within this block's rows
    float4 v = ((const float4*)(t + (long)row * DD))[k4];
    float* dstp = &sA[wv * (16 * 132) + r * 132 + k4 * 4];
    dstp[0] = v.x; dstp[1] = v.y; dstp[2] = v.z; dstp[3] = v.w;
  }
  __syncthreads();

  const int half = lane >> 4;
  const int l16 = lane & 15;
  const float* Abase = &sA[wv * (16 * 132) + l16 * 132 + half * 2];

  for (int ct = 0; ct < 8; ++ct) {
    v8f acc = {};
    const float* Bbase = &sW[(half * 2) * DD + ct * 16 + l16];
#pragma unroll
    for (int k0 = 0; k0 < 32; ++k0) {
      v2f a, bb;
      a.x = Abase[k0 * 4];
      a.y = Abase[k0 * 4 + 1];
      bb.x = Bbase[(k0 * 4) * DD];
      bb.y = Bbase[(k0 * 4 + 1) * DD];
      acc = __builtin_amdgcn_wmma_f32_16x16x4_f32(false, a, false, bb, (short)0, acc,
                                                  false, false);
    }
    const int col = ct * 16 + l16;
    const float bias = b[col];
#pragma unroll
    for (int i = 0; i < 8; ++i) {
      int row = rowbase + i + half * 8;
      if (row < N) {
        float v = acc[i] + bias;
        if (resid) v += resid[(long)row * DD + col];
        t[(long)row * DD + col] = v;
      }
    }
  }
}

// ---------------- batchnorm ----------------
__global__ __launch_bounds__(128) void k_stats(const float* __restrict__ h,
                                               float* __restrict__ stats, int N) {
  int c = threadIdx.x;
  int r0 = blockIdx.x * 512;
  int rend = r0 + 512;
  if (rend > N) rend = N;
  float sum = 0.f, sq = 0.f;
  for (int r = r0; r < rend; ++r) {
    float v = h[(long)r * DD + c];
    sum += v;
    sq += v * v;
  }
  atomAddF(&stats[c], sum);
  atomAddF(&stats[DD + c], sq);
}

__global__ __launch_bounds__(128) void k_stats_final(float* __restrict__ stats, int N) {
  int c = threadIdx.x;
  float invN = 1.0f / (float)N;
  float mu = stats[c] * invN;
  float var = stats[DD + c] * invN - mu * mu;
  stats[2 * DD + c] = mu;
  stats[3 * DD + c] = rsqrtf(var + 1e-5f);
}

__global__ __launch_bounds__(256) void k_bnrelu(const float* __restrict__ h,
                                                const float* __restrict__ stats,
                                                const float* __restrict__ gamma,
                                                const float* __restrict__ beta,
                                                float* __restrict__ out, long nd) {
  long i = (long)blockIdx.x * blockDim.x + threadIdx.x;
  if (i >= nd) return;
  int c = (int)(i & (DD - 1));
  float v = (h[i] - stats[2 * DD + c]) * stats[3 * DD + c] * gamma[c] + beta[c];
  out[i] = fmaxf(v, 0.f);
}

// ---------------- pooling + head ----------------
__global__ __launch_bounds__(256) void k_pool(const float* __restrict__ h,
                                              const int* __restrict__ batch,
                                              float* __restrict__ hg, float* __restrict__ cnt,
                                              long nd) {
  long i = (long)blockIdx.x * blockDim.x + threadIdx.x;
  if (i >= nd) return;
  long n = i >> 7;
  int c = (int)(i & (DD - 1));
  int g = batch[n];
  atomAddF(&hg[(long)g * DD + c], h[i]);
  if (c == 0) atomAddF(&cnt[g], 1.0f);
}

__global__ __launch_bounds__(128) void k_final(const float* __restrict__ hg,
                                               const float* __restrict__ cnt,
                                               const float* __restrict__ fw,
                                               const float* __restrict__ fb,
                                               const float* __restrict__ betaMix,
                                               const float* __restrict__ rf,
                                               float* __restrict__ out) {
  __shared__ float red[128];
  int g = blockIdx.x, c = threadIdx.x;
  float cm = fmaxf(cnt[g], 1.0f);
  red[c] = hg[(long)g * DD + c] / cm * fw[c];
  __syncthreads();
  for (int off = 64; off > 0; off >>= 1) {
    if (c < off) red[c] += red[c + off];
    __syncthreads();
  }
  if (c == 0) {
    float z = red[0] + fb[0];
    float sg = 1.f / (1.f + __expf(-z));
    float bm = betaMix[0];
    out[g] = (1.f - bm) * sg + bm * rf[g];
  }
}

extern "C" void kernel_launch(void* const* d_in, const int* in_sizes, int n_in,
                              void* d_out, int out_size, void* d_ws, size_t ws_size,
                              hipStream_t stream) {
  const int* x = (const int*)d_in[0];
  const int* ei = (const int*)d_in[1];
  const int* ea = (const int*)d_in[2];
  const int* batch = (const int*)d_in[3];
  const float* rf = (const float*)d_in[4];
  const float* aemb = (const float*)d_in[5];
  const float* bemb = (const float*)d_in[6];
  const float* cw = (const float*)d_in[7];
  const float* cb = (const float*)d_in[8];
  const float* ts = (const float*)d_in[9];
  const float* gam = (const float*)d_in[10];
  const float* bet = (const float*)d_in[11];
  const float* fw = (const float*)d_in[12];
  const float* fb = (const float*)d_in[13];
  const float* betaMix = (const float*)d_in[14];

  const int N = in_sizes[3];
  const int E = in_sizes[2] / 3;
  const int G = in_sizes[4];
  const int L = in_sizes[9];
  const long nd = (long)N * DD;

  float* ws = (float*)d_ws;
  float* hcur = ws;                // residual stream
  float* h2 = ws + nd;             // bn-relu output
  float* tcur = ws + 2 * nd;       // conv input / in-place gemm output
  float* stats = ws + 3 * nd;      // 4*128 (sum, sumsq, mu, inv)
  float* hg = stats + 4 * DD;      // G*128 pooled
  float* cnt = hg + (long)G * DD;  // G
  int* deg = (int*)(cnt + G);      // N
  int* fill = deg + N;             // N
  int* rowptr = fill + N;          // N+1
  int* eidx = rowptr + N + 1;      // E

  const int blkND = (int)((nd + 255) / 256);
  const int blkE = (E + 255) / 256;
  const int blkN32 = (int)(((long)N * 32 + 255) / 256);  // wave per node
  const int blkStats = (N + 511) / 512;
  const int blkGemm = (N + 127) / 128;

  // CSR of incoming edges (deterministic work, rebuilt every call)
  k_zero_int<<<(2 * N + 255) / 256, 256, 0, stream>>>(deg, 2L * N);  // deg + fill
  k_count<<<blkE, 256, 0, stream>>>(ei, deg, E);
  k_scan<<<1, 1024, 0, stream>>>(deg, rowptr, N);
  k_scatter<<<blkE, 256, 0, stream>>>(ei, rowptr, fill, eidx, E);

  // initial node embeddings
  k_atom<<<blkND, 256, 0, stream>>>(x, aemb, hcur, nd);

  for (int l = 0; l < L; ++l) {
    const float* hin;
    if (l == 0) {
      hin = hcur;
    } else {
      k_zero<<<1, 256, 0, stream>>>(stats, 2 * DD);
      k_stats<<<blkStats, 128, 0, stream>>>(hcur, stats, N);
      k_stats_final<<<1, 128, 0, stream>>>(stats, N);
      k_bnrelu<<<blkND, 256, 0, stream>>>(hcur, stats, gam + (long)(l - 1) * DD,
                                          bet + (long)(l - 1) * DD, h2, nd);
      hin = h2;
    }
    k_aggr<<<blkN32, 256, 0, stream>>>(ei, ea, rowptr, eidx, bemb, hin, ts, l, tcur, N, E);
    k_gemm<<<blkGemm, 256, 0, stream>>>(tcur, cw + (long)l * DD * DD, cb + (long)l * DD,
                                        (l == 0) ? nullptr : hcur, N);
    float* tmp = hcur;
    hcur = tcur;
    tcur = tmp;
  }

  // final BN + ReLU
  k_zero<<<1, 256, 0, stream>>>(stats, 2 * DD);
  k_stats<<<blkStats, 128, 0, stream>>>(hcur, stats, N);
  k_stats_final<<<1, 128, 0, stream>>>(stats, N);
  k_bnrelu<<<blkND, 256, 0, stream>>>(hcur, stats, gam + (long)(L - 1) * DD,
                                      bet + (long)(L - 1) * DD, h2, nd);

  // mean pooling + head
  k_zero<<<(int)(((long)G * (DD + 1) + 255) / 256), 256, 0, stream>>>(hg, (long)G * (DD + 1));
  k_pool<<<blkND, 256, 0, stream>>>(h2, batch, hg, cnt, nd);
  k_final<<<G, 128, 0, stream>>>(hg, cnt, fw, fb, betaMix, rf, (float*)d_out);
}